// XEmbedding_38001870635381
// MI455X (gfx1250) — compile-verified
//
#include <hip/hip_runtime.h>
#include <math.h>

#define N_NODES   100000
#define N_EDGES   640000
#define EMBED_DIM 28
#define NODE_DIM  128
#define NUM_BASIS 20
#define CUTOFF    5.0f

typedef __attribute__((ext_vector_type(2))) float v2f;
typedef __attribute__((ext_vector_type(4))) float v4f;
typedef __attribute__((ext_vector_type(8))) float v8f;

// ---------------------------------------------------------------------------
// Kernel 1: x_scalar = embed_table[at_no] @ node_w + node_b
// f32 WMMA 16x16x4, one wave per 16-row tile, K=28 = 7 steps of 4.
// ---------------------------------------------------------------------------
__global__ __launch_bounds__(256) void node_embed_wmma(
    const int*   __restrict__ at_no,
    const float* __restrict__ embed_table,
    const float* __restrict__ node_w,
    const float* __restrict__ node_b,
    float*       __restrict__ x_out)
{
    const int lane = threadIdx.x & 31;
    const int wave = blockIdx.x * (blockDim.x >> 5) + (threadIdx.x >> 5);
    const int m0   = wave << 4;
    if (m0 >= N_NODES) return;              // wave-uniform: EXEC stays all-1s

    // A fragment layout (32-bit A 16x4): lanes 0-15 -> rows M=lane, K = 4k+{0,1}
    //                                    lanes 16-31 -> rows M=lane-16, K = 4k+{2,3}
    const int   mrow  = m0 + (lane & 15);
    const int   type  = at_no[mrow];
    const float* arow = embed_table + type * EMBED_DIM;
    const int   khalf = (lane >> 4) << 1;   // 0 or 2

    v2f afrag[7];
#pragma unroll
    for (int ks = 0; ks < 7; ++ks) {
        const float2 a = *(const float2*)(arow + 4 * ks + khalf); // 8B aligned
        afrag[ks].x = a.x;
        afrag[ks].y = a.y;
    }

    const int nlo   = lane & 15;
    const int mbase = m0 + ((lane >> 4) << 3);   // D: VGPR r -> M = r (+8 hi half)

#pragma unroll
    for (int nt = 0; nt < 8; ++nt) {
        const int ncol = nt * 16 + nlo;
        v8f acc = {};
#pragma unroll
        for (int ks = 0; ks < 7; ++ks) {
            const int k0 = 4 * ks + khalf;
            v2f b;                                   // B 4x16: rows striped like A
            b.x = node_w[k0 * NODE_DIM + ncol];
            b.y = node_w[(k0 + 1) * NODE_DIM + ncol];
            acc = __builtin_amdgcn_wmma_f32_16x16x4_f32(
                false, afrag[ks], false, b, (short)0, acc, false, false);
        }
        const float bias = node_b[ncol];
#pragma unroll
        for (int r = 0; r < 8; ++r) {
            __builtin_nontemporal_store(acc[r] + bias,
                                        x_out + (size_t)(mbase + r) * NODE_DIM + ncol);
        }
    }
}

// ---------------------------------------------------------------------------
// Kernel 2: per-edge RBF / cutoff / tiled spherical harmonics. One wave/edge.
// ---------------------------------------------------------------------------
__device__ __forceinline__ float sel3(int t, float a, float b, float c) {
    return t == 0 ? a : (t == 1 ? b : c);
}
__device__ __forceinline__ float sel5(int t, float a, float b, float c, float d, float e) {
    return t < 2 ? (t == 0 ? a : b) : (t == 2 ? c : (t == 3 ? d : e));
}

__global__ __launch_bounds__(256) void edge_features(
    const float* __restrict__ pos,
    const int*   __restrict__ edge_index,
    const float* __restrict__ freq,
    float*       __restrict__ rbf_out,
    float*       __restrict__ fcut_out,
    float*       __restrict__ rsh_out)
{
    const int lane = threadIdx.x & 31;
    int e = blockIdx.x * (blockDim.x >> 5) + (threadIdx.x >> 5);
    if (e >= N_EDGES) return;
    e = __builtin_amdgcn_readfirstlane(e);   // provably uniform -> scalar gathers

    const int i = edge_index[e];
    const int j = edge_index[N_EDGES + e];

    // axis permutation [1,2,0] applied to pos before differencing
    const float vx = pos[i * 3 + 1] - pos[j * 3 + 1];
    const float vy = pos[i * 3 + 2] - pos[j * 3 + 2];
    const float vz = pos[i * 3 + 0] - pos[j * 3 + 0];
    const float dist = sqrtf(vx * vx + vy * vy + vz * vz);
    const float inv  = 1.0f / dist;
    const float x = vx * inv, y = vy * inv, z = vz * inv;

    // RBF: sqrt(2/CUTOFF) * sin(freq*dist)/dist   (lanes 0..19)
    if (lane < NUM_BASIS) {
        const float s = 0.63245553203367586f;            // sqrt(2/5)
        __builtin_nontemporal_store(s * sinf(freq[lane] * dist) * inv,
                                    rbf_out + (size_t)e * NUM_BASIS + lane);
    }
    // fcut = 1 - 10 r^3 + 15 r^4 - 6 r^5   (p = 3)
    if (lane == 0) {
        const float r  = dist * (1.0f / CUTOFF);
        const float r3 = r * r * r;
        __builtin_nontemporal_store(1.0f - 10.0f * r3 + 15.0f * r3 * r - 6.0f * r3 * r * r,
                                    fcut_out + e);
    }

    const float s3  = 1.7320508075688772f;
    const float s5  = 2.2360679774997896f;
    const float s15 = 3.8729833462074170f;
    const float a0 = s3 * x, a1 = s3 * y, a2 = s3 * z;
    const float b0 = s15 * x * z;
    const float b1 = s15 * x * y;
    const float b2 = s5 * (y * y - 0.5f * (x * x + z * z));
    const float b3 = s15 * y * z;
    const float b4 = 0.5f * s15 * (z * z - x * x);

    float* rrow = rsh_out + (size_t)e * 480;

    // Segment 0: cols [0,128) = sh0 tiled x128 -> all 1.0 (32 x B128 NT stores)
    {
        v4f ones = {1.0f, 1.0f, 1.0f, 1.0f};
        __builtin_nontemporal_store(ones, (v4f*)(rrow + lane * 4));
    }
    // Segment 1: cols [128,320) = [a0,a1,a2] tiled x64 (48 float4)
#pragma unroll
    for (int rnd = 0; rnd < 2; ++rnd) {
        const int q = lane + rnd * 32;
        if (q < 48) {
            const int o = q * 4;
            v4f v;
            v.x = sel3(o % 3, a0, a1, a2);
            v.y = sel3((o + 1) % 3, a0, a1, a2);
            v.z = sel3((o + 2) % 3, a0, a1, a2);
            v.w = sel3((o + 3) % 3, a0, a1, a2);
            __builtin_nontemporal_store(v, (v4f*)(rrow + 128 + o));
        }
    }
    // Segment 2: cols [320,480) = [b0..b4] tiled x32 (40 float4)
#pragma unroll
    for (int rnd = 0; rnd < 2; ++rnd) {
        const int q = lane + rnd * 32;
        if (q < 40) {
            const int o = q * 4;
            v4f v;
            v.x = sel5(o % 5, b0, b1, b2, b3, b4);
            v.y = sel5((o + 1) % 5, b0, b1, b2, b3, b4);
            v.z = sel5((o + 2) % 5, b0, b1, b2, b3, b4);
            v.w = sel5((o + 3) % 5, b0, b1, b2, b3, b4);
            __builtin_nontemporal_store(v, (v4f*)(rrow + 320 + o));
        }
    }
}

// ---------------------------------------------------------------------------
extern "C" void kernel_launch(void* const* d_in, const int* in_sizes, int n_in,
                              void* d_out, int out_size, void* d_ws, size_t ws_size,
                              hipStream_t stream)
{
    const int*   at_no       = (const int*)d_in[0];
    const float* pos         = (const float*)d_in[1];
    const int*   edge_index  = (const int*)d_in[2];
    const float* embed_table = (const float*)d_in[3];
    const float* node_w      = (const float*)d_in[4];
    const float* node_b      = (const float*)d_in[5];
    const float* freq        = (const float*)d_in[6];

    float* out      = (float*)d_out;
    float* x_scalar = out;                                        // 100000*128
    float* rbf      = x_scalar + (size_t)N_NODES * NODE_DIM;      // 640000*20
    float* fcut     = rbf + (size_t)N_EDGES * NUM_BASIS;          // 640000
    float* rsh      = fcut + (size_t)N_EDGES;                     // 640000*480

    // Kernel 1: 6250 M-tiles, 8 waves (256 threads) per block
    const int waves1  = N_NODES / 16;                 // 6250
    const int blocks1 = (waves1 + 7) / 8;             // 782
    node_embed_wmma<<<blocks1, 256, 0, stream>>>(at_no, embed_table, node_w,
                                                 node_b, x_scalar);

    // Kernel 2: one wave per edge, 8 waves per block
    const int blocks2 = N_EDGES / 8;                  // 80000
    edge_features<<<blocks2, 256, 0, stream>>>(pos, edge_index, freq,
                                               rbf, fcut, rsh);
}